// ChamferLoss_89352499626133
// MI455X (gfx1250) — compile-verified
//
#include <hip/hip_runtime.h>

typedef __attribute__((ext_vector_type(2))) float v2f;
typedef __attribute__((ext_vector_type(8))) float v8f;

#define B_ 8
#define H_ 3
#define P_ 4096
#define ROW_BLOCKS 32          // 4096 / 128 row-blocks per (b,h)
#define ROWS_PER_WAVE 16
#define THREADS 256            // 8 waves of 32

// Dynamic LDS layout: [ float4 yst[P_*2] | unsigned colMinLds[P_] ]
#define YST_BYTES   (P_ * 2 * 16)            // 128 KB
#define LDS_BYTES   (YST_BYTES + P_ * 4)     // 144 KB of 320 KB/WGP

#define BIGF 3.0e38f

// ---------------------------------------------------------------------------
// Workspace init: column-min array -> +inf bits, accumulators -> 0
// ---------------------------------------------------------------------------
__global__ void init_ws_kernel(unsigned int* __restrict__ colMin,
                               float* __restrict__ accum) {
    int i = blockIdx.x * blockDim.x + threadIdx.x;
    if (i < B_ * H_ * P_) colMin[i] = 0x7F800000u;   // +inf
    if (i < 2) accum[i] = 0.0f;
}

// ---------------------------------------------------------------------------
// Regularizer: sum_b sqrt(|| N N^T - I ||_F^2), N = row-normalized normals.
// ---------------------------------------------------------------------------
__global__ void reg_kernel(const float* __restrict__ planes,
                           float* __restrict__ accum) {
    int lane = threadIdx.x;
    float reg = 0.0f;
    if (lane < B_) {
        float n[H_][3];
        for (int h = 0; h < H_; ++h) {
            const float* p = planes + (lane * H_ + h) * 4;
            float nx = p[0], ny = p[1], nz = p[2];
            float inv = rsqrtf(nx * nx + ny * ny + nz * nz);
            n[h][0] = nx * inv; n[h][1] = ny * inv; n[h][2] = nz * inv;
        }
        float s = 0.0f;
        for (int i = 0; i < H_; ++i)
            for (int j = 0; j < H_; ++j) {
                float g = n[i][0]*n[j][0] + n[i][1]*n[j][1] + n[i][2]*n[j][2]
                        - (i == j ? 1.0f : 0.0f);
                s += g * g;
            }
        reg = sqrtf(s);
    }
    for (int m = 1; m < 32; m <<= 1) reg += __shfl_xor(reg, m, 32);
    if (lane == 0) atomicAdd(&accum[1], 25.0f * reg);
}

// ---------------------------------------------------------------------------
// Chamfer core. Block = 8 waves on one (b,h,rowBlk): 128 rows vs all 4096
// columns. The reflected cloud is staged ONCE per block into LDS as two
// float4 per point: [q*2+0] = (ryx, ryy, y2, 0)   (B-operand K0/K1 + y2)
//                   [q*2+1] = (ryz, 0,   y2, 0)   (B-operand K2/K3 + y2)
// so each lane fetches its WMMA B pair AND the C-preload y2 with one
// aligned LDS load. Distance tile via V_WMMA_F32_16X16X4_F32:
//   A = -2x (16x4, K3=0), B = y (4x16), C = x2_row + y2_col  ->  D = dist^2
// Clamp to [0, inf) via v_med3 (raw-operand intrinsic: no canonicalize ops).
// ---------------------------------------------------------------------------
__global__ void __launch_bounds__(THREADS)
chamfer_kernel(const float* __restrict__ planes,
               const float* __restrict__ pts,
               unsigned int* __restrict__ colMinG,
               float* __restrict__ accum) {
    extern __shared__ unsigned char smem[];
    float4*       yst       = (float4*)smem;                     // P_*2 entries
    unsigned int* colMinLds = (unsigned int*)(smem + YST_BYTES); // P_ entries

    const int tid  = threadIdx.x;
    const int lane = tid & 31;
    const int wave = tid >> 5;
    const int m    = lane & 15;                     // row/col within tile
    const int half = lane >> 4;                     // 0: K0/K1, 1: K2/K3

    const int blk    = blockIdx.x;
    const int rowBlk = blk & (ROW_BLOCKS - 1);
    const int bh     = blk / ROW_BLOCKS;
    const int h      = bh % H_;
    const int b      = bh / H_;

    // plane -> normalized normal + raw offset (matches reference)
    const float* pl = planes + (b * H_ + h) * 4;
    float nx = pl[0], ny = pl[1], nz = pl[2], off = pl[3];
    float inv = rsqrtf(nx * nx + ny * ny + nz * nz);
    nx *= inv; ny *= inv; nz *= inv;

    const float* xb = pts + (size_t)b * P_ * 3;

    // ---- stage reflected cloud + col-min array in LDS (once per block) ----
    for (int q = tid; q < P_; q += THREADS) {
        float qx = xb[q * 3 + 0];
        float qy = xb[q * 3 + 1];
        float qz = xb[q * 3 + 2];
        float dpl = nx * qx + ny * qy + nz * qz + off;
        float ryx = qx - 2.0f * dpl * nx;
        float ryy = qy - 2.0f * dpl * ny;
        float ryz = qz - 2.0f * dpl * nz;
        float y2  = ryx * ryx + ryy * ryy + ryz * ryz;
        yst[q * 2 + 0] = make_float4(ryx, ryy, y2, 0.0f);
        yst[q * 2 + 1] = make_float4(ryz, 0.0f, y2, 0.0f);
        colMinLds[q] = 0x7F800000u;
    }
    __syncthreads();

    // ---- A operand: this wave's 16 rows, -2x folded in ----
    const int rowbase = rowBlk * 128 + wave * ROWS_PER_WAVE;
    const int prow    = rowbase + m;
    float px = xb[prow * 3 + 0];
    float py = xb[prow * 3 + 1];
    float pz = xb[prow * 3 + 2];
    float x2m = px * px + py * py + pz * pz;        // lane holds |x_row|^2

    v2f a;
    a.x = half ? (-2.0f * pz) : (-2.0f * px);       // K2 : K0
    a.y = half ? 0.0f         : (-2.0f * py);       // K3 : K1

    // x2 broadcast per C-layout: VGPR r, lane l -> row r + 8*(l>=16)
    float x2h[8];
#pragma unroll
    for (int r = 0; r < 8; ++r)
        x2h[r] = __shfl(x2m, r + half * 8, 32);     // ds_bpermute

    float rmin[8];
#pragma unroll
    for (int r = 0; r < 8; ++r) rmin[r] = BIGF;

    for (int t = 0; t < P_ / 16; ++t) {
        const int q = t * 16 + m;
        float4 v = yst[q * 2 + half];               // one aligned LDS load

        v8f c;
#pragma unroll
        for (int r = 0; r < 8; ++r) c[r] = x2h[r] + v.z;

        v2f bb; bb.x = v.x; bb.y = v.y;

        // D = A*B + C : full 16x16 squared-distance tile
        c = __builtin_amdgcn_wmma_f32_16x16x4_f32(
                false, a, false, bb, (short)0, c, false, false);

        float cm = BIGF;
#pragma unroll
        for (int r = 0; r < 8; ++r) {
            // clamp to >= 0 with v_med3 (no canonicalize of WMMA output)
            float d = __builtin_amdgcn_fmed3f(c[r], 0.0f, BIGF);
            rmin[r] = fminf(rmin[r], d);
            cm = fminf(cm, d);
        }
        // both halves min their 8-row partial into the same column slot:
        // combined effect is the full 16-row min, no shfl, no branch.
        atomicMin(&colMinLds[q], __float_as_uint(cm));
    }

    // ---- row mins: reduce across the 16 column-lanes (masks stay in-half) --
#pragma unroll
    for (int r = 0; r < 8; ++r) {
        float v = rmin[r];
        v = fminf(v, __shfl_xor(v, 1, 32));
        v = fminf(v, __shfl_xor(v, 2, 32));
        v = fminf(v, __shfl_xor(v, 4, 32));
        v = fminf(v, __shfl_xor(v, 8, 32));
        rmin[r] = v;
    }
    float s = 0.0f;
#pragma unroll
    for (int r = 0; r < 8; ++r) s += rmin[r];       // rows base..+7 / +8..+15
    s += __shfl_xor(s, 16, 32);
    if (lane == 0) atomicAdd(&accum[0], s);

    // ---- publish block-local column mins (one global atomic per column) ----
    __syncthreads();
    unsigned int* cg = colMinG + (size_t)(b * H_ + h) * P_;
    for (int i = tid; i < P_; i += THREADS)
        atomicMin(&cg[i], colMinLds[i]);
}

// ---------------------------------------------------------------------------
// Final: sum column mins, combine with row-min sum and regularizer.
// refl = (sum rowmins + sum colmins) / (B*P);  out = refl + 25*reg
// ---------------------------------------------------------------------------
__global__ void finalize_kernel(const unsigned int* __restrict__ colMinG,
                                const float* __restrict__ accum,
                                float* __restrict__ out) {
    __shared__ float red[THREADS];
    float s = 0.0f;
    for (int i = threadIdx.x; i < B_ * H_ * P_; i += THREADS)
        s += __uint_as_float(colMinG[i]);
    red[threadIdx.x] = s;
    __syncthreads();
    for (int k = THREADS / 2; k > 0; k >>= 1) {
        if ((int)threadIdx.x < k) red[threadIdx.x] += red[threadIdx.x + k];
        __syncthreads();
    }
    if (threadIdx.x == 0)
        out[0] = (accum[0] + red[0]) / (float)(B_ * P_) + accum[1];
}

// ---------------------------------------------------------------------------
extern "C" void kernel_launch(void* const* d_in, const int* in_sizes, int n_in,
                              void* d_out, int out_size, void* d_ws, size_t ws_size,
                              hipStream_t stream) {
    const float* planes = (const float*)d_in[0];   // (8,3,4) f32
    const float* pts    = (const float*)d_in[1];   // (8,4096,3) f32
    // d_in[2], d_in[3] (voxel/cp grids) are zeros and unused by the reference.

    float*        accum  = (float*)d_ws;                          // [0]=rowsum,[1]=25*reg
    unsigned int* colMin = (unsigned int*)((char*)d_ws + 16);     // B*H*P uints

    const int nCol = B_ * H_ * P_;
    init_ws_kernel<<<(nCol + THREADS - 1) / THREADS, THREADS, 0, stream>>>(colMin, accum);
    reg_kernel<<<1, 32, 0, stream>>>(planes, accum);
    chamfer_kernel<<<B_ * H_ * ROW_BLOCKS, THREADS, LDS_BYTES, stream>>>(planes, pts, colMin, accum);
    finalize_kernel<<<1, THREADS, 0, stream>>>(colMin, accum, (float*)d_out);
}